// MarketRegimeEncoder_15109694948218
// MI455X (gfx1250) — compile-verified
//
#include <hip/hip_runtime.h>
#include <math.h>

typedef float v2f __attribute__((ext_vector_type(2)));
typedef float v8f __attribute__((ext_vector_type(8)));

#define EPSF    1e-8f
#define LN_EPSF 1e-5f
#define B_ROWS  8192
#define S_LEN   4096
#define HID     256   // 4 encoders * 64
#define HEAD_H  64

// ---------------------------------------------------------------------------
// Kernel 1: per-row feature extraction + tiny encoder + LayerNorm.
// One 256-thread block per batch row; row staged in LDS (16 KB).
// Writes regime_features (B,256) directly in the reference's e-major layout.
// ---------------------------------------------------------------------------
__global__ __launch_bounds__(256) void regime_features_kernel(
    const float* __restrict__ prices,    // (B, S)
    const float* __restrict__ enc_w,     // (4, 4, 64)
    const float* __restrict__ enc_b,     // (4, 64)
    const float* __restrict__ enc_g,     // (4, 64)
    const float* __restrict__ enc_beta,  // (4, 64)
    float* __restrict__ rf)              // (B, 256)
{
    __shared__ float sp[S_LEN];          // 16 KB row buffer
    __shared__ float r1[256];
    __shared__ float r2[256];
    __shared__ float scomb[4];
    __shared__ float smean;
    __shared__ float lnbuf[256];

    const int tid = threadIdx.x;
    const int b   = blockIdx.x;
    const float* row = prices + (size_t)b * S_LEN;

    for (int j = tid; j < S_LEN; j += 256) sp[j] = row[j];
    __syncthreads();

    const int i0 = tid * 16;             // contiguous 16-element chunk / thread

    // ---- pass 1: row mean ----
    float lsum = 0.f;
    for (int i = i0; i < i0 + 16; ++i) lsum += sp[i];
    r1[tid] = lsum;
    __syncthreads();
    for (int s = 128; s > 0; s >>= 1) {
        if (tid < s) r1[tid] += r1[tid + s];
        __syncthreads();
    }
    if (tid == 0) smean = r1[0] * (1.0f / (float)S_LEN);
    __syncthreads();
    const float mean = smean;

    // ---- pass 2: rolling-SMA trend signal + centered variance ----
    float winsum = 0.f;
    int start = i0 - 19; if (start < 0) start = 0;
    for (int j = start; j <= i0; ++j) winsum += sp[j];

    float trend = 0.f, var_acc = 0.f;
    for (int i = i0; i < i0 + 16; ++i) {
        if (i != i0) {
            winsum += sp[i];
            if (i - 20 >= 0) winsum -= sp[i - 20];
        }
        const float cnt = (float)((i + 1 < 20) ? (i + 1) : 20);
        const float sma = winsum / cnt;
        const float p   = sp[i];
        trend += (p - sma) / (sma + EPSF);
        const float d = p - mean;
        var_acc += d * d;
    }
    __syncthreads();
    r1[tid] = trend;
    r2[tid] = var_acc;
    __syncthreads();
    for (int s = 128; s > 0; s >>= 1) {
        if (tid < s) { r1[tid] += r1[tid + s]; r2[tid] += r2[tid + s]; }
        __syncthreads();
    }
    if (tid == 0) {
        const float f_trend = r1[0] * (1.0f / (float)S_LEN);
        const float p_last  = sp[S_LEN - 1];
        const float p_m10   = sp[S_LEN - 10];
        const float f_mom   = (p_last - p_m10) / (p_m10 + EPSF);
        const float f_mr    = (p_last - mean) / (mean + EPSF);
        const float var1    = r2[0] / (float)(S_LEN - 1);   // ddof=1
        const float f_cyc   = sqrtf(var1) / (mean + EPSF);
        scomb[0] = f_trend; scomb[1] = f_mom; scomb[2] = f_mr; scomb[3] = f_cyc;
    }
    __syncthreads();

    // ---- encoder: tid -> (e = tid/64, o = tid%64) ----
    const int e = tid >> 6;
    const int o = tid & 63;
    float pre = enc_b[e * 64 + o];
#pragma unroll
    for (int i = 0; i < 4; ++i)
        pre += scomb[i] * enc_w[e * 256 + i * 64 + o];
    const float x = pre > 0.f ? pre : 0.f;   // relu
    lnbuf[tid] = x;
    __syncthreads();

    // ---- LayerNorm over each group of 64 ----
    const int gb = e * 64;
    float mu = 0.f;
    for (int j = 0; j < 64; ++j) mu += lnbuf[gb + j];
    mu *= (1.0f / 64.0f);
    float v = 0.f;
    for (int j = 0; j < 64; ++j) { const float d = lnbuf[gb + j] - mu; v += d * d; }
    v *= (1.0f / 64.0f);
    const float outv = (x - mu) / sqrtf(v + LN_EPSF) * enc_g[gb + o] + enc_beta[gb + o];
    rf[(size_t)b * HID + tid] = outv;
}

// ---------------------------------------------------------------------------
// Kernel 2: head MLP via V_WMMA_F32_16X16X4_F32 (exact f32 matrix math).
// 256 threads = 8 waves; each wave owns a 16-row tile of h = relu(RF@W1+b1).
// A (16x4 f32): lanes 0-15 hold K={0,1}, lanes 16-31 hold K={2,3} (2 VGPRs).
// B (4x16 f32): lane-half selects K-half, VGPR index selects K within half.
// ---------------------------------------------------------------------------
__global__ __launch_bounds__(256) void regime_head_kernel(
    const float* __restrict__ rf,        // (B, 256)
    const float* __restrict__ w1,        // (256, 64)
    const float* __restrict__ b1,        // (64)
    const float* __restrict__ w2,        // (64, 4)
    const float* __restrict__ b2,        // (4)
    float* __restrict__ probs)           // (B, 4)
{
    __shared__ float hbuf[128 * HEAD_H]; // 32 KB h staging (128 rows per block)

    const int tid     = threadIdx.x;
    const int lane    = tid & 31;        // wave32
    const int wave    = tid >> 5;
    const int rowbase = blockIdx.x * 128 + wave * 16;
    const int m16     = lane & 15;
    const int koff    = (lane >= 16) ? 2 : 0;
    const float* arow = rf + (size_t)(rowbase + m16) * HID;

    v8f acc0 = {0.f,0.f,0.f,0.f,0.f,0.f,0.f,0.f};
    v8f acc1 = acc0, acc2 = acc0, acc3 = acc0;

    for (int kk = 0; kk < HID; kk += 4) {
        v2f a;
        a.x = arow[kk + koff];
        a.y = arow[kk + koff + 1];
        const float* w1r0 = w1 + (size_t)(kk + koff) * HEAD_H + m16;
        const float* w1r1 = w1r0 + HEAD_H;
        v2f bm;
        bm.x = w1r0[0];  bm.y = w1r1[0];
        acc0 = __builtin_amdgcn_wmma_f32_16x16x4_f32(false, a, false, bm, (short)0, acc0, false, false);
        bm.x = w1r0[16]; bm.y = w1r1[16];
        acc1 = __builtin_amdgcn_wmma_f32_16x16x4_f32(false, a, false, bm, (short)0, acc1, false, false);
        bm.x = w1r0[32]; bm.y = w1r1[32];
        acc2 = __builtin_amdgcn_wmma_f32_16x16x4_f32(false, a, false, bm, (short)0, acc2, false, false);
        bm.x = w1r0[48]; bm.y = w1r1[48];
        acc3 = __builtin_amdgcn_wmma_f32_16x16x4_f32(false, a, false, bm, (short)0, acc3, false, false);
    }

    // C/D layout: VGPR r -> M=r (lanes 0-15) / M=r+8 (lanes 16-31); N = lane%16 within tile.
    const int mlo = (lane < 16) ? 0 : 8;
#pragma unroll
    for (int r = 0; r < 8; ++r) {
        const int mloc = wave * 16 + mlo + r;
        float h0 = acc0[r] + b1[ 0 + m16]; h0 = fmaxf(h0, 0.f);
        float h1 = acc1[r] + b1[16 + m16]; h1 = fmaxf(h1, 0.f);
        float h2 = acc2[r] + b1[32 + m16]; h2 = fmaxf(h2, 0.f);
        float h3 = acc3[r] + b1[48 + m16]; h3 = fmaxf(h3, 0.f);
        hbuf[mloc * HEAD_H +  0 + m16] = h0;
        hbuf[mloc * HEAD_H + 16 + m16] = h1;
        hbuf[mloc * HEAD_H + 32 + m16] = h2;
        hbuf[mloc * HEAD_H + 48 + m16] = h3;
    }
    __syncthreads();

    // 64x4 projection + softmax; one thread per row.
    if (tid < 128) {
        const float* hr = hbuf + tid * HEAD_H;
        float l0 = b2[0], l1 = b2[1], l2 = b2[2], l3 = b2[3];
        for (int n = 0; n < HEAD_H; ++n) {
            const float h = hr[n];
            l0 += h * w2[n * 4 + 0];
            l1 += h * w2[n * 4 + 1];
            l2 += h * w2[n * 4 + 2];
            l3 += h * w2[n * 4 + 3];
        }
        const float mx = fmaxf(fmaxf(l0, l1), fmaxf(l2, l3));
        const float e0 = expf(l0 - mx), e1 = expf(l1 - mx);
        const float e2 = expf(l2 - mx), e3 = expf(l3 - mx);
        const float inv = 1.0f / (e0 + e1 + e2 + e3);
        float* po = probs + (size_t)(blockIdx.x * 128 + tid) * 4;
        po[0] = e0 * inv; po[1] = e1 * inv; po[2] = e2 * inv; po[3] = e3 * inv;
    }
}

extern "C" void kernel_launch(void* const* d_in, const int* in_sizes, int n_in,
                              void* d_out, int out_size, void* d_ws, size_t ws_size,
                              hipStream_t stream) {
    (void)in_sizes; (void)n_in; (void)out_size; (void)d_ws; (void)ws_size;
    const float* prices   = (const float*)d_in[0];
    const float* enc_w    = (const float*)d_in[1];
    const float* enc_b    = (const float*)d_in[2];
    const float* enc_g    = (const float*)d_in[3];
    const float* enc_beta = (const float*)d_in[4];
    const float* head_w1  = (const float*)d_in[5];
    const float* head_b1  = (const float*)d_in[6];
    const float* head_w2  = (const float*)d_in[7];
    const float* head_b2  = (const float*)d_in[8];

    float* rf    = (float*)d_out;                       // (B, 256)
    float* probs = rf + (size_t)B_ROWS * HID;           // (B, 4)

    regime_features_kernel<<<B_ROWS, 256, 0, stream>>>(
        prices, enc_w, enc_b, enc_g, enc_beta, rf);
    regime_head_kernel<<<B_ROWS / 128, 256, 0, stream>>>(
        rf, head_w1, head_b1, head_w2, head_b2, probs);
}